// TRCategorical_79388175499500
// MI455X (gfx1250) — compile-verified
//
#include <hip/hip_runtime.h>
#include <hip/hip_bf16.h>

typedef __attribute__((ext_vector_type(2))) float v2f;
typedef __attribute__((ext_vector_type(8))) float v8f;

#define K_CORES 12
#define D_DIM   8192
#define R_DIM   16
#define MAT     256          // R*R
#define LDS_RS  18           // padded LDS row stride (dwords): even (b64-aligned pairs),
                             // 18 coprime-enough with 64 banks -> conflict-free rows

// ---------------------------------------------------------------------------
// Kernel 1: cores = softplus(log_cores), elementwise over K*D*R*R floats.
// ---------------------------------------------------------------------------
__global__ __launch_bounds__(256) void softplus_kernel(
    const float* __restrict__ log_cores, float* __restrict__ cores, int n)
{
    int i = blockIdx.x * 256 + threadIdx.x;
    if (i < n) {
        float x = log_cores[i];
        // softplus(x) = max(x,0) + log1p(exp(-|x|))
        cores[i] = fmaxf(x, 0.0f) + log1pf(__expf(-fabsf(x)));
    }
}

// ---------------------------------------------------------------------------
// Kernel 2: norm[k] = sum_d cores[k][d]  ->  [K][256]
// One block per k; thread t owns matrix element t, strides over D.
// ---------------------------------------------------------------------------
__global__ __launch_bounds__(256) void norm_sum_kernel(
    const float* __restrict__ cores, float* __restrict__ norm)
{
    int k = blockIdx.x;
    int t = threadIdx.x;
    const float* base = cores + (size_t)k * D_DIM * MAT + t;
    float s = 0.0f;
    for (int d = 0; d < D_DIM; ++d)
        s += base[(size_t)d * MAT];
    norm[k * MAT + t] = s;
}

// ---------------------------------------------------------------------------
// Kernel 3: chain the 12 norm matrices, logZ = log(trace(prod)). One block.
// ---------------------------------------------------------------------------
__global__ __launch_bounds__(256) void norm_chain_kernel(
    const float* __restrict__ norm, float* __restrict__ logZ)
{
    __shared__ float A[MAT];
    __shared__ float Bm[MAT];
    __shared__ float Cm[MAT];
    __shared__ float tr[R_DIM];
    int t = threadIdx.x;
    int m = t >> 4, n = t & 15;

    A[t] = norm[t];
    __syncthreads();
    for (int k = 1; k < K_CORES; ++k) {
        Bm[t] = norm[k * MAT + t];
        __syncthreads();
        float acc = 0.0f;
        #pragma unroll
        for (int j = 0; j < R_DIM; ++j)
            acc = __builtin_fmaf(A[m * 16 + j], Bm[j * 16 + n], acc);
        Cm[t] = acc;
        __syncthreads();
        A[t] = Cm[t];
        __syncthreads();
    }
    if (m == n) tr[m] = A[t];
    __syncthreads();
    if (t == 0) {
        float s = 0.0f;
        #pragma unroll
        for (int i = 0; i < R_DIM; ++i) s += tr[i];
        *logZ = __logf(s);
    }
}

// ---------------------------------------------------------------------------
// Kernel 4: main batched chain. One sample per wave32, 8 waves per block.
//   acc (C/D layout, v8f) = core0[idx0] @ core1[idx1]   (A from global, B gathered)
//   for k = 2..11: acc = acc @ corek[idxk]  via LDS round-trip (D->A relayout)
//   out[b] = log(trace(acc)) - logZ
// ---------------------------------------------------------------------------
__global__ __launch_bounds__(256) void tr_chain_kernel(
    const int*   __restrict__ index,
    const float* __restrict__ cores,
    const float* __restrict__ logZp,
    float*       __restrict__ out)
{
    __shared__ __align__(16) float lds[8][R_DIM * LDS_RS];

    const int lane = threadIdx.x & 31;
    const int wave = threadIdx.x >> 5;
    const int b    = blockIdx.x * 8 + wave;
    float* wlds = lds[wave];

    const int half = lane >> 4;   // 0: lanes 0-15, 1: lanes 16-31
    const int l16  = lane & 15;

    const int* idxp = index + b * K_CORES;

    // ---- step 0+1 fused: A = core0 (A layout from global), B = core1 (B layout)
    const float* c0 = cores + ((size_t)0 * D_DIM + idxp[0]) * MAT;
    const float* c1 = cores + ((size_t)1 * D_DIM + idxp[1]) * MAT;

    v2f a[4];
    #pragma unroll
    for (int kk = 0; kk < 4; ++kk) {
        // lane holds row M=l16; V0 gets K=4kk+2*half, V1 gets K+1 (consecutive)
        a[kk] = *(const v2f*)(c0 + l16 * R_DIM + 4 * kk + 2 * half);
    }
    v2f bm[4];
    #pragma unroll
    for (int kk = 0; kk < 4; ++kk) {
        // B chunk kk (4x16): V0 row 4kk+2*half, V1 row 4kk+1+2*half, col l16
        bm[kk].x = c1[(4 * kk + 2 * half)     * R_DIM + l16];
        bm[kk].y = c1[(4 * kk + 1 + 2 * half) * R_DIM + l16];
    }
    v8f acc = {};
    #pragma unroll
    for (int kk = 0; kk < 4; ++kk)
        acc = __builtin_amdgcn_wmma_f32_16x16x4_f32(
            false, a[kk], false, bm[kk], (short)0, acc, false, false);

    // ---- steps k = 2..11: D-layout acc -> LDS -> A-layout, gather B, 4 WMMAs
    for (int k = 2; k < K_CORES; ++k) {
        const float* ck = cores + ((size_t)k * D_DIM + idxp[k]) * MAT;

        // spill acc: VGPR r holds (M = r + 8*half, N = l16)
        #pragma unroll
        for (int r = 0; r < 8; ++r)
            wlds[(r + 8 * half) * LDS_RS + l16] = acc[r];

        // reload in A layout: lane row = l16, K pair = 4kk+2*half (b64, even offset)
        v2f aa[4];
        #pragma unroll
        for (int kk = 0; kk < 4; ++kk)
            aa[kk] = *(const v2f*)(wlds + l16 * LDS_RS + 4 * kk + 2 * half);

        v2f bb[4];
        #pragma unroll
        for (int kk = 0; kk < 4; ++kk) {
            bb[kk].x = ck[(4 * kk + 2 * half)     * R_DIM + l16];
            bb[kk].y = ck[(4 * kk + 1 + 2 * half) * R_DIM + l16];
        }

        v8f nacc = {};
        #pragma unroll
        for (int kk = 0; kk < 4; ++kk)
            nacc = __builtin_amdgcn_wmma_f32_16x16x4_f32(
                false, aa[kk], false, bb[kk], (short)0, nacc, false, false);
        acc = nacc;
    }

    // ---- trace: diag element r lives in lane r (half 0) and lane r+24 (half 1)
    float c = 0.0f;
    #pragma unroll
    for (int r = 0; r < 8; ++r)
        c += (lane == r || lane == r + 24) ? acc[r] : 0.0f;

    #pragma unroll
    for (int off = 16; off > 0; off >>= 1)
        c += __shfl_xor(c, off, 32);

    if (lane == 0)
        out[b] = __logf(c) - logZp[0];
}

// ---------------------------------------------------------------------------
extern "C" void kernel_launch(void* const* d_in, const int* in_sizes, int n_in,
                              void* d_out, int out_size, void* d_ws, size_t ws_size,
                              hipStream_t stream)
{
    const int*   index     = (const int*)d_in[0];      // [B, K] int32
    const float* log_cores = (const float*)d_in[1];    // [K, D, R, R] f32
    float*       out       = (float*)d_out;            // [B] f32

    const int B = in_sizes[0] / K_CORES;
    const int n_core = K_CORES * D_DIM * MAT;          // 25,165,824 floats (~100.7 MB)

    // workspace layout: [softplus'd cores | norm sums | logZ scalar]
    float* ws_cores = (float*)d_ws;
    float* ws_norm  = ws_cores + (size_t)n_core;
    float* ws_logZ  = ws_norm + K_CORES * MAT;

    softplus_kernel  <<<n_core / 256, 256, 0, stream>>>(log_cores, ws_cores, n_core);
    norm_sum_kernel  <<<K_CORES,      256, 0, stream>>>(ws_cores, ws_norm);
    norm_chain_kernel<<<1,            256, 0, stream>>>(ws_norm, ws_logZ);
    tr_chain_kernel  <<<B / 8,        256, 0, stream>>>(index, ws_cores, ws_logZ, out);
}